// PropNet_30133490549597
// MI455X (gfx1250) — compile-verified
//
#include <hip/hip_runtime.h>

typedef __attribute__((ext_vector_type(2))) float v2f;
typedef __attribute__((ext_vector_type(8))) float v8f;

#define B_    8192
#define N_    5
#define R_    (B_ * N_)     // 40960 entity rows
#define DIN_  198
#define KP_   200           // K padded to multiple of 4
#define H_    256
#define DOUT_ 128
#define CB_   1280          // 5 stacked 256-wide channel blocks: [Ps|Pr|Qr|Pw|Qw]
#define BN_EPS_ 1e-5f

// ---------------- workspace layout (floats) ----------------
// requires ws_size >= ~287 MB
static constexpr long WS_F     = 0;                          // R_*KP_
static constexpr long WS_WA    = WS_F    + (long)R_ * KP_;   // KP_*CB_
static constexpr long WS_Y     = WS_WA   + (long)KP_ * CB_;  // R_*CB_
static constexpr long WS_STAT  = WS_Y    + (long)R_ * CB_;   // 1536: [mu_s|rs_s|mu_r|rs_r|mu_w|rs_w]
static constexpr long WS_ALPHA = WS_STAT + 1536;             // 256
static constexpr long WS_SS    = WS_ALPHA + 256;             // 256 (BN scale, self)
static constexpr long WS_TS    = WS_SS    + 256;             // 256 (BN shift, self)
static constexpr long WS_SCAL  = WS_TS    + 256;             // 32: [beta, logit_pad]
static constexpr long WS_WATT  = WS_SCAL  + 32;              // R_*8 attention weights
static constexpr long WS_WSUM  = WS_WATT  + (long)R_ * 8;    // R_
static constexpr long WS_HAGG  = WS_WSUM  + R_;              // R_*H_

// ---------------- feature assembly: F[R x 200] = concat(...), padded ----------------
__global__ __launch_bounds__(256) void k_build_feat(
    const float* __restrict__ zp, const float* __restrict__ zw,
    const float* __restrict__ zwhat, const float* __restrict__ zd,
    const float* __restrict__ st, float* __restrict__ F) {
  long idx = (long)blockIdx.x * blockDim.x + threadIdx.x;
  if (idx >= (long)R_ * KP_) return;
  int row = (int)(idx / KP_), c = (int)(idx % KP_);
  float v = 0.f;
  if (c < 1)        v = zp[row];
  else if (c < 5)   v = zw[row * 4 + (c - 1)];
  else if (c < 69)  v = zwhat[row * 64 + (c - 5)];
  else if (c < 70)  v = zd[row];
  else if (c < 198) v = st[row * 128 + (c - 70)];
  F[idx] = v;
}

// ---------------- algebraically folded weight stack: Waug[200 x 1280] ----------------
// blk0 = W1s; blk1 = W1r_top  (rows 3,4 += W1r rows 201,202)   -> P_r from feat_i
// blk2 = W1r_bot (rows 201,202 negated)                        -> Q_r from feat_j
// blk3/blk4: same construction from W1w.
__global__ __launch_bounds__(256) void k_build_w(
    const float* __restrict__ W1s, const float* __restrict__ W1r,
    const float* __restrict__ W1w, float* __restrict__ Wa) {
  int idx = blockIdx.x * blockDim.x + threadIdx.x;
  if (idx >= KP_ * CB_) return;
  int k = idx / CB_, c = idx % CB_;
  int blk = c >> 8, cc = c & 255;
  float v = 0.f;
  if (k < DIN_) {
    bool sh = (k == 3 || k == 4);   // z_shift components
    if (blk == 0)      v = W1s[k * H_ + cc];
    else if (blk == 1) { v = W1r[k * H_ + cc]; if (sh) v += W1r[(k + DIN_) * H_ + cc]; }
    else if (blk == 2) { v = W1r[(k + DIN_) * H_ + cc]; if (sh) v = -v; }
    else if (blk == 3) { v = W1w[k * H_ + cc]; if (sh) v += W1w[(k + DIN_) * H_ + cc]; }
    else               { v = W1w[(k + DIN_) * H_ + cc]; if (sh) v = -v; }
  }
  Wa[idx] = v;
}

// ---------------- GEMM1: Y[R x 1280] = F[R x 200] @ Waug[200 x 1280] ----------------
// one wave per 16x64 C tile; fp32 WMMA 16x16x4, K loop of 50 steps.
__global__ __launch_bounds__(32) void k_gemm1(
    const float* __restrict__ F, const float* __restrict__ Wa,
    float* __restrict__ Y) {
  const int m0 = blockIdx.x * 16, n0 = blockIdx.y * 64;
  const int lane = threadIdx.x, r = lane & 15, hi = lane >> 4;
  v8f acc[4] = {};
  const float* frow = F + (long)(m0 + r) * KP_;
  for (int k0 = 0; k0 < KP_; k0 += 4) {
    const int ka = k0 + 2 * hi;
    v2f a; a.x = frow[ka]; a.y = frow[ka + 1];
    const float* w0 = Wa + (long)ka * CB_ + n0 + r;
    const float* w1 = w0 + CB_;
#pragma unroll
    for (int t = 0; t < 4; t++) {
      v2f bf; bf.x = w0[16 * t]; bf.y = w1[16 * t];
      acc[t] = __builtin_amdgcn_wmma_f32_16x16x4_f32(
          false, a, false, bf, (short)0, acc[t], false, false);
    }
  }
#pragma unroll
  for (int i = 0; i < 8; i++) {
    float* yr = Y + (long)(m0 + i + 8 * hi) * CB_ + n0 + r;
#pragma unroll
    for (int t = 0; t < 4; t++) yr[16 * t] = acc[t][i];
  }
}

// ---------------- BN stats, self MLP: over R rows of relu(Ps + b1s) ----------------
__global__ __launch_bounds__(256) void k_stats_s(
    const float* __restrict__ Y, const float* __restrict__ b1,
    float* __restrict__ statOut) {
  int c = blockIdx.x;
  float bb = b1[c], s = 0.f, s2 = 0.f;
  for (int row = threadIdx.x; row < R_; row += 256) {
    float h = fmaxf(Y[(long)row * CB_ + c] + bb, 0.f);
    s += h; s2 += h * h;
  }
  __shared__ float sh[256], sh2[256];
  sh[threadIdx.x] = s; sh2[threadIdx.x] = s2;
  __syncthreads();
  for (int o = 128; o > 0; o >>= 1) {
    if (threadIdx.x < o) { sh[threadIdx.x] += sh[threadIdx.x + o]; sh2[threadIdx.x] += sh2[threadIdx.x + o]; }
    __syncthreads();
  }
  if (threadIdx.x == 0) {
    float mu = sh[0] / (float)R_;
    float var = sh2[0] / (float)R_ - mu * mu;
    statOut[c] = mu;
    statOut[256 + c] = rsqrtf(var + BN_EPS_);
  }
}

// ---------------- BN stats over factored pair rows relu(P_i + Q_j + b1) ----------------
// addPad: also counts R_ all-zero input rows (h = relu(b1)) for the weight MLP.
__global__ __launch_bounds__(256) void k_stats_pair(
    const float* __restrict__ Y, const float* __restrict__ b1,
    int offP, int offQ, int addPad, float* __restrict__ statOut) {
  int c = blockIdx.x;
  float bb = b1[c], s = 0.f, s2 = 0.f;
  for (int b = threadIdx.x; b < B_; b += 256) {
    float p[N_], q[N_];
#pragma unroll
    for (int i = 0; i < N_; i++) {
      p[i] = Y[(long)(b * N_ + i) * CB_ + offP + c];
      q[i] = Y[(long)(b * N_ + i) * CB_ + offQ + c];
    }
#pragma unroll
    for (int i = 0; i < N_; i++)
#pragma unroll
      for (int j = 0; j < N_; j++) {
        float h = fmaxf(p[i] + q[j] + bb, 0.f);
        s += h; s2 += h * h;
      }
  }
  __shared__ float sh[256], sh2[256];
  sh[threadIdx.x] = s; sh2[threadIdx.x] = s2;
  __syncthreads();
  for (int o = 128; o > 0; o >>= 1) {
    if (threadIdx.x < o) { sh[threadIdx.x] += sh[threadIdx.x + o]; sh2[threadIdx.x] += sh2[threadIdx.x + o]; }
    __syncthreads();
  }
  if (threadIdx.x == 0) {
    float tot = sh[0], tot2 = sh2[0];
    float cnt = (float)((long)R_ * N_);
    if (addPad) {
      float hp = fmaxf(bb, 0.f);
      tot += (float)R_ * hp; tot2 += (float)R_ * hp * hp;
      cnt += (float)R_;
    }
    float mu = tot / cnt;
    float var = tot2 / cnt - mu * mu;
    statOut[c] = mu;
    statOut[256 + c] = rsqrtf(var + BN_EPS_);
  }
}

// ---------------- fold BN(self) into per-channel scale/shift ----------------
__global__ __launch_bounds__(256) void k_svec(
    const float* __restrict__ stat, const float* __restrict__ gs,
    const float* __restrict__ bes, float* __restrict__ ss, float* __restrict__ ts) {
  int c = threadIdx.x;
  float mu = stat[c], rs = stat[256 + c];
  float s = rs * gs[c];
  ss[c] = s;
  ts[c] = bes[c] - mu * s;
}

// ---------------- fold BN(w) + W2w into alpha[c], beta, pad logit ----------------
__global__ __launch_bounds__(256) void k_wvec(
    const float* __restrict__ statw, const float* __restrict__ gw,
    const float* __restrict__ bew, const float* __restrict__ W2w,
    const float* __restrict__ b2w, const float* __restrict__ b1w,
    float* __restrict__ alpha, float* __restrict__ scal) {
  int c = threadIdx.x;
  float mu = statw[c], rs = statw[256 + c];
  float sg = rs * gw[c];
  float a = sg * W2w[c];
  alpha[c] = a;
  float betac = (bew[c] - mu * sg) * W2w[c];
  float padc = fmaxf(b1w[c], 0.f) * a;
  __shared__ float sb[256], sp[256];
  sb[c] = betac; sp[c] = padc;
  __syncthreads();
  for (int o = 128; o > 0; o >>= 1) {
    if (c < o) { sb[c] += sb[c + o]; sp[c] += sp[c + o]; }
    __syncthreads();
  }
  if (c == 0) {
    float beta = sb[0] + b2w[0];
    scal[0] = beta;          // logit offset
    scal[1] = sp[0] + beta;  // pad-column logit
  }
}

// ---------------- attention: logits -> softmax -> z_pres -> diag mask -> renorm ----
// one wave per (b,i); 8 waves/block.
__global__ __launch_bounds__(256) void k_attn(
    const float* __restrict__ Y, const float* __restrict__ b1w,
    const float* __restrict__ alpha, const float* __restrict__ scal,
    const float* __restrict__ zpres, float* __restrict__ Watt,
    float* __restrict__ wsum, float* __restrict__ outw) {
  const int wid = threadIdx.x >> 5, lane = threadIdx.x & 31;
  const int row = blockIdx.x * 8 + wid;
  const int b = row / N_, i = row % N_;
  float preg[8], areg[8];
#pragma unroll
  for (int t = 0; t < 8; t++) {
    int c = lane + 32 * t;
    preg[t] = Y[(long)row * CB_ + 768 + c] + b1w[c];
    areg[t] = alpha[c];
  }
  const float beta = scal[0];
  float lg[6];
  for (int j = 0; j < N_; j++) {
    const float* q = Y + (long)(b * N_ + j) * CB_ + 1024;
    float acc = 0.f;
#pragma unroll
    for (int t = 0; t < 8; t++) {
      int c = lane + 32 * t;
      acc = fmaf(fmaxf(preg[t] + q[c], 0.f), areg[t], acc);
    }
#pragma unroll
    for (int o = 16; o > 0; o >>= 1) acc += __shfl_xor(acc, o, 32);
    lg[j] = acc + beta;
  }
  lg[5] = scal[1];
  float m = lg[0];
#pragma unroll
  for (int j = 1; j < 6; j++) m = fmaxf(m, lg[j]);
  float w[6], S = 0.f;
#pragma unroll
  for (int j = 0; j < 6; j++) { w[j] = __expf(lg[j] - m); S += w[j]; }
  float invS = 1.f / S;
#pragma unroll
  for (int j = 0; j < 6; j++) w[j] *= invS;          // true softmax
#pragma unroll
  for (int j = 0; j < N_; j++) w[j] *= zpres[b * N_ + j];
  w[5] = 0.f;                                        // zp pad column
#pragma unroll
  for (int j = 0; j < N_; j++) if (j == i) w[j] = 0.f;   // diag mask
  float T = 0.f;
#pragma unroll
  for (int j = 0; j < 6; j++) T += w[j];
  float inv = 1.f / (T + 1e-4f);
#pragma unroll
  for (int j = 0; j < 6; j++) w[j] *= inv;
  if (lane == 0) {
#pragma unroll
    for (int j = 0; j < 6; j++) Watt[row * 8 + j] = w[j];
#pragma unroll
    for (int j = 0; j < N_; j++) outw[(long)b * 25 + i * 5 + j] = w[j];
    wsum[row] = T * inv;
  }
}

// ---------------- aggregate Hagg[b,i] = sum_j w_ij * BN(relu(Pr_i + Qr_j + b1r)) ----
__global__ __launch_bounds__(256) void k_agg(
    const float* __restrict__ Y, const float* __restrict__ b1r,
    const float* __restrict__ statr, const float* __restrict__ gr,
    const float* __restrict__ ber, const float* __restrict__ Watt,
    float* __restrict__ Hagg) {
  const int row = blockIdx.x, b = row / N_, c = threadIdx.x;
  float mu = statr[c], rs = statr[256 + c];
  float sc = rs * gr[c];
  float tc = ber[c] - mu * sc;
  float p = Y[(long)row * CB_ + 256 + c] + b1r[c];
  float acc = 0.f;
#pragma unroll
  for (int j = 0; j < N_; j++) {
    float wv = Watt[row * 8 + j];
    float h = fmaxf(p + Y[(long)(b * N_ + j) * CB_ + 512 + c], 0.f);
    acc = fmaf(wv, fmaf(h, sc, tc), acc);
  }
  Hagg[(long)row * H_ + c] = acc;
}

// ---------------- GEMM2 (fused): out = BN(relu(Ps+b1s))@W2s + Hagg@W2r + b2s + wsum*b2r
__global__ __launch_bounds__(32) void k_gemm2(
    const float* __restrict__ Y, const float* __restrict__ Hagg,
    const float* __restrict__ W2s, const float* __restrict__ W2r,
    const float* __restrict__ b1s, const float* __restrict__ ss,
    const float* __restrict__ ts, const float* __restrict__ b2s,
    const float* __restrict__ b2r, const float* __restrict__ wsum,
    float* __restrict__ out) {
  const int m0 = blockIdx.x * 16, n0 = blockIdx.y * 64;
  const int lane = threadIdx.x, r = lane & 15, hi = lane >> 4;
  const int arow = m0 + r;
  v8f acc[4] = {};
  // pass 1: self branch (BN folded into A-fragment load)
  for (int k0 = 0; k0 < H_; k0 += 4) {
    const int ka = k0 + 2 * hi;
    v2f a;
    a.x = fmaf(fmaxf(Y[(long)arow * CB_ + ka] + b1s[ka], 0.f), ss[ka], ts[ka]);
    a.y = fmaf(fmaxf(Y[(long)arow * CB_ + ka + 1] + b1s[ka + 1], 0.f), ss[ka + 1], ts[ka + 1]);
    const float* w0 = W2s + (long)ka * DOUT_ + n0 + r;
    const float* w1 = w0 + DOUT_;
#pragma unroll
    for (int t = 0; t < 4; t++) {
      v2f bf; bf.x = w0[16 * t]; bf.y = w1[16 * t];
      acc[t] = __builtin_amdgcn_wmma_f32_16x16x4_f32(
          false, a, false, bf, (short)0, acc[t], false, false);
    }
  }
  // pass 2: relational branch
  for (int k0 = 0; k0 < H_; k0 += 4) {
    const int ka = k0 + 2 * hi;
    v2f a; a.x = Hagg[(long)arow * H_ + ka]; a.y = Hagg[(long)arow * H_ + ka + 1];
    const float* w0 = W2r + (long)ka * DOUT_ + n0 + r;
    const float* w1 = w0 + DOUT_;
#pragma unroll
    for (int t = 0; t < 4; t++) {
      v2f bf; bf.x = w0[16 * t]; bf.y = w1[16 * t];
      acc[t] = __builtin_amdgcn_wmma_f32_16x16x4_f32(
          false, a, false, bf, (short)0, acc[t], false, false);
    }
  }
#pragma unroll
  for (int i = 0; i < 8; i++) {
    int rr = m0 + i + 8 * hi;
    float wsr = wsum[rr];
#pragma unroll
    for (int t = 0; t < 4; t++) {
      int col = n0 + 16 * t + r;
      out[(long)rr * DOUT_ + col] = acc[t][i] + b2s[col] + wsr * b2r[col];
    }
  }
}

extern "C" void kernel_launch(void* const* d_in, const int* in_sizes, int n_in,
                              void* d_out, int out_size, void* d_ws, size_t ws_size,
                              hipStream_t stream) {
  const float* zp    = (const float*)d_in[0];
  const float* zw    = (const float*)d_in[1];
  const float* zwhat = (const float*)d_in[2];
  const float* zd    = (const float*)d_in[3];
  const float* st    = (const float*)d_in[4];
  const float* W1s = (const float*)d_in[5];
  const float* b1s = (const float*)d_in[6];
  const float* gs  = (const float*)d_in[7];
  const float* bes = (const float*)d_in[8];
  const float* W2s = (const float*)d_in[9];
  const float* b2s = (const float*)d_in[10];
  const float* W1r = (const float*)d_in[11];
  const float* b1r = (const float*)d_in[12];
  const float* gr  = (const float*)d_in[13];
  const float* ber = (const float*)d_in[14];
  const float* W2r = (const float*)d_in[15];
  const float* b2r = (const float*)d_in[16];
  const float* W1w = (const float*)d_in[17];
  const float* b1w = (const float*)d_in[18];
  const float* gw  = (const float*)d_in[19];
  const float* bew = (const float*)d_in[20];
  const float* W2w = (const float*)d_in[21];
  const float* b2w = (const float*)d_in[22];

  float* wsf   = (float*)d_ws;
  float* F     = wsf + WS_F;
  float* Wa    = wsf + WS_WA;
  float* Y     = wsf + WS_Y;
  float* stat  = wsf + WS_STAT;
  float* alpha = wsf + WS_ALPHA;
  float* ssv   = wsf + WS_SS;
  float* tsv   = wsf + WS_TS;
  float* scal  = wsf + WS_SCAL;
  float* Watt  = wsf + WS_WATT;
  float* wsm   = wsf + WS_WSUM;
  float* Hagg  = wsf + WS_HAGG;

  float* out_state = (float*)d_out;               // (B,N,128)
  float* out_w     = out_state + (long)R_ * DOUT_; // (B,5,5,1)

  k_build_feat<<<((long)R_ * KP_ + 255) / 256, 256, 0, stream>>>(zp, zw, zwhat, zd, st, F);
  k_build_w<<<(KP_ * CB_ + 255) / 256, 256, 0, stream>>>(W1s, W1r, W1w, Wa);
  k_gemm1<<<dim3(R_ / 16, CB_ / 64), 32, 0, stream>>>(F, Wa, Y);
  k_stats_s<<<256, 256, 0, stream>>>(Y, b1s, stat);
  k_stats_pair<<<256, 256, 0, stream>>>(Y, b1r, 256, 512, 0, stat + 512);
  k_stats_pair<<<256, 256, 0, stream>>>(Y, b1w, 768, 1024, 1, stat + 1024);
  k_svec<<<1, 256, 0, stream>>>(stat, gs, bes, ssv, tsv);
  k_wvec<<<1, 256, 0, stream>>>(stat + 1024, gw, bew, W2w, b2w, b1w, alpha, scal);
  k_attn<<<R_ / 8, 256, 0, stream>>>(Y, b1w, alpha, scal, zp, Watt, wsm, out_w);
  k_agg<<<R_, 256, 0, stream>>>(Y, b1r, stat + 512, gr, ber, Watt, Hagg);
  k_gemm2<<<dim3(R_ / 16, DOUT_ / 64), 32, 0, stream>>>(
      Y, Hagg, W2s, W2r, b1s, ssv, tsv, b2s, b2r, wsm, out_state);
}